// DetectionPostprocess_63479616634937
// MI455X (gfx1250) — compile-verified
//
#include <hip/hip_runtime.h>
#include <hip/hip_bf16.h>
#include <stdint.h>

// Detection postprocess for MI455X (gfx1250, wave32).
// One workgroup (256 thr = 8 waves) per batch element.
//  - TDM tensor_load_to_lds (unconditional) streams the batch's 13824 logits into LDS
//  - iterative block argmax with per-thread cached local max -> top-60
//    (sorted desc, tie -> smaller index, matches jax.lax.top_k)
//  - v_wmma_f32_16x16x4_f32 computes the 64x64 (padded) vol_i + vol_j matrix
//    as the rank-2 product [vol|1] x [1;vol^T] for the IoU denominator
//  - thread 0 runs the sequential greedy NMS (<=20 keeps) from LDS

#define NB     256
#define NANCH  13824      // 24*24*24, == 54*256 exactly
#define NTOPK  60
#define STRIDE 4.0f       // 96/24
#define THRESH 0.15f
#define NMS_THRESH 0.05f
#define NMS_KEEP 20
#define VOL_THRESH 180.0f
#define NEG_INF (-3.0e38f)

typedef float v2f __attribute__((ext_vector_type(2)));
typedef float v8f __attribute__((ext_vector_type(8)));
typedef unsigned int u32x4 __attribute__((ext_vector_type(4)));
typedef int i32x4 __attribute__((ext_vector_type(4)));
typedef int i32x8 __attribute__((ext_vector_type(8)));

#if defined(__has_builtin)
#if __has_builtin(__builtin_amdgcn_s_wait_tensorcnt)
#define WAIT_TENSORCNT0() __builtin_amdgcn_s_wait_tensorcnt(0)
#endif
#endif
#ifndef WAIT_TENSORCNT0
#define WAIT_TENSORCNT0() asm volatile("s_wait_tensorcnt 0x0" ::: "memory")
#endif

__global__ __launch_bounds__(256)
void det_postprocess_kernel(const float* __restrict__ cls,
                            const float* __restrict__ shp,
                            const float* __restrict__ off,
                            float* __restrict__ out) {
    __shared__ float s_logits[NANCH];     // 55296 B; reused as volsum(4096f)+iou(3600f)
    __shared__ float s_red_val[256];
    __shared__ int   s_red_idx[256];
    __shared__ int   s_topk_idx[NTOPK];
    __shared__ float s_topk_logit[NTOPK];
    __shared__ float s_det[NTOPK * 8];
    __shared__ float s_score[NTOPK];
    __shared__ float s_vol[64];
    __shared__ float s_out[NTOPK * 8];

    const int tid  = threadIdx.x;
    const int b    = blockIdx.x;
    const float* cls_b = cls + (size_t)b * NANCH;

    // ---------- Phase 0: batch logits -> LDS via Tensor Data Mover ----------
    if (tid < 32) {   // single wave issues the TDM op (EXEC-independent, SGPR descriptor)
        unsigned long long ga = (unsigned long long)(uintptr_t)cls_b;
        unsigned lds_off = (unsigned)(uintptr_t)s_logits;

        u32x4 g0;
        g0.x = 1u;                                   // count=1 valid descriptor
        g0.y = lds_off;                              // lds_addr (bytes)
        g0.z = (unsigned)(ga & 0xFFFFFFFFu);         // global_addr[31:0]
        g0.w = (unsigned)((ga >> 32) & 0x1FFFFFFu)   // global_addr[56:32]
             | (2u << 30);                           // type=2 ("image")

        i32x8 g1;
        g1[0] = (2 << 16);                           // data_size=4B; no multicast/pad
        g1[1] = (int)((unsigned)NANCH << 16);        // tensor_dim0[15:0] @ [31:16]
        g1[2] = (int)(((unsigned)NANCH >> 16)        // tensor_dim0[31:16]
                      | (1u << 16));                 // tensor_dim1 = 1
        g1[3] = (int)((unsigned)NANCH << 16);        // tile_dim0 @ [127:112]
        g1[4] = 1;                                   // tile_dim1 = 1, tile_dim2 = 0
        g1[5] = NANCH;                               // tensor_dim0_stride[31:0]
        g1[6] = (int)((unsigned)NANCH << 16);        // stride0 hi=0 | stride1[15:0]
        g1[7] = 0;                                   // stride1 hi

        i32x4 z4 = {0, 0, 0, 0};
#if defined(__clang_major__) && (__clang_major__ >= 23)
        i32x8 z8 = {0, 0, 0, 0, 0, 0, 0, 0};
        __builtin_amdgcn_tensor_load_to_lds(g0, g1, z4, z4, z8, 0);
#else
        __builtin_amdgcn_tensor_load_to_lds(g0, g1, z4, z4, 0);
#endif
        WAIT_TENSORCNT0();
    }
    __syncthreads();

    // ---------- Phase 1: top-60, per-thread cached local max ----------
    // Element k is owned by thread k%256 (stride-256 stripes of 54 elements).
    float lbest = NEG_INF;
    int   lbidx = 0x7FFFFFFF;
    for (int k = tid; k < NANCH; k += 256) {
        float v = s_logits[k];
        if (v > lbest) { lbest = v; lbidx = k; }    // ascending scan keeps first on tie
    }

    for (int it = 0; it < NTOPK; ++it) {
        s_red_val[tid] = lbest;
        s_red_idx[tid] = lbidx;
        __syncthreads();
        // LDS tree reduce 256 -> 32
        for (int s = 128; s >= 32; s >>= 1) {
            if (tid < s) {
                float v1 = s_red_val[tid],     v2 = s_red_val[tid + s];
                int   i1 = s_red_idx[tid],     i2 = s_red_idx[tid + s];
                if (v2 > v1 || (v2 == v1 && i2 < i1)) {
                    s_red_val[tid] = v2; s_red_idx[tid] = i2;
                }
            }
            __syncthreads();
        }
        // in-wave reduce 32 -> 1 (wave32 shuffles, no barriers)
        if (tid < 32) {
            float v = s_red_val[tid];
            int   i = s_red_idx[tid];
            #pragma unroll
            for (int d = 16; d > 0; d >>= 1) {
                float v2 = __shfl_xor(v, d, 32);
                int   i2 = __shfl_xor(i, d, 32);
                if (v2 > v || (v2 == v && i2 < i)) { v = v2; i = i2; }
            }
            if (tid == 0) { s_red_val[0] = v; s_red_idx[0] = i; }
        }
        __syncthreads();

        const int   w  = s_red_idx[0];
        const float wv = s_red_val[0];
        if (tid == 0) {
            s_topk_idx[it]   = w;
            s_topk_logit[it] = wv;
        }
        // Only the owner's local max is invalidated: suppress + rescan its stripe.
        if ((w & 255) == tid) {
            s_logits[w] = NEG_INF;
            lbest = NEG_INF;
            lbidx = 0x7FFFFFFF;
            for (int k = tid; k < NANCH; k += 256) {
                float v = s_logits[k];
                if (v > lbest) { lbest = v; lbidx = k; }
            }
        }
        __syncthreads();
    }

    // ---------- Phase 2: decode boxes for the 60 candidates ----------
    if (tid < NTOPK) {
        int   idx   = s_topk_idx[tid];
        float score = 1.0f / (1.0f + expf(-s_topk_logit[tid]));
        int z = idx / 576;
        int r = idx - z * 576;
        int y = r / 24;
        int x = r - y * 24;
        const float* ofb = off + (size_t)b * 3 * NANCH;
        const float* shb = shp + (size_t)b * 3 * NANCH;
        float cz = ((float)z + ofb[0 * NANCH + idx]) * STRIDE;
        float cy = ((float)y + ofb[1 * NANCH + idx]) * STRIDE;
        float cx = ((float)x + ofb[2 * NANCH + idx]) * STRIDE;
        float sz = shb[0 * NANCH + idx];
        float sy = shb[1 * NANCH + idx];
        float sx = shb[2 * NANCH + idx];
        s_det[tid * 8 + 0] = 1.0f;
        s_det[tid * 8 + 1] = score;
        s_det[tid * 8 + 2] = cz;
        s_det[tid * 8 + 3] = cy;
        s_det[tid * 8 + 4] = cx;
        s_det[tid * 8 + 5] = sz;
        s_det[tid * 8 + 6] = sy;
        s_det[tid * 8 + 7] = sx;
        s_score[tid] = score;
        s_vol[tid]   = sz * sy * sx;
    }
    if (tid >= NTOPK && tid < 64) s_vol[tid] = 0.0f;
    for (int p = tid; p < NTOPK * 8; p += 256) s_out[p] = -1.0f;
    __syncthreads();

    // ---------- Phase 3: volsum[i][j] = vol_i + vol_j via WMMA ----------
    // Rank-2 product: A(16x4) = [vol | 1 | 0 | 0], B(4x16) = [1 ; vol^T ; 0 ; 0].
    float* s_volsum = s_logits;          // 64*64 floats (logits are dead now)
    float* s_iou    = s_logits + 4096;   // 60*60 floats
    {
        const int lane = tid & 31;
        const int wave = tid >> 5;
        const int r    = lane & 15;
        const int half = lane >> 4;      // 0: K=0,1  1: K=2,3 (zero)
        for (int t = wave * 2; t < wave * 2 + 2; ++t) {
            const int mt = t >> 2, nt = t & 3;
            v2f a, bm;
            a.x  = half ? 0.0f : s_vol[mt * 16 + r];  // K0 = vol_m
            a.y  = half ? 0.0f : 1.0f;                // K1 = 1
            bm.x = half ? 0.0f : 1.0f;                // K0 row = ones
            bm.y = half ? 0.0f : s_vol[nt * 16 + r];  // K1 row = vol_n
            v8f c = {};
            c = __builtin_amdgcn_wmma_f32_16x16x4_f32(
                    false, a, false, bm, (short)0, c, false, false);
            const int row0 = half * 8;
            const int col  = r;
            #pragma unroll
            for (int rr = 0; rr < 8; ++rr)
                s_volsum[(mt * 16 + row0 + rr) * 64 + nt * 16 + col] = c[rr];
        }
    }
    __syncthreads();

    // ---------- Phase 4: 60x60 IoU ----------
    for (int p = tid; p < NTOPK * NTOPK; p += 256) {
        int i = p / NTOPK, j = p - i * NTOPK;
        float inter = 1.0f;
        #pragma unroll
        for (int c = 0; c < 3; ++c) {
            float ci = s_det[i * 8 + 2 + c], si = s_det[i * 8 + 5 + c];
            float cj = s_det[j * 8 + 2 + c], sj = s_det[j * 8 + 5 + c];
            float lo = fmaxf(ci - 0.5f * si, cj - 0.5f * sj);
            float hi = fminf(ci + 0.5f * si, cj + 0.5f * sj);
            inter *= fmaxf(hi - lo, 0.0f);
        }
        s_iou[p] = inter / (s_volsum[i * 64 + j] - inter + 1e-8f);
    }
    __syncthreads();

    // ---------- Phase 5: sequential greedy NMS (exact reference recurrence) ----------
    if (tid == 0) {
        int keep = 0;
        int sup[NTOPK];
        for (int i = 0; i < NTOPK; ++i) sup[i] = (s_score[i] > THRESH) ? 0 : 1;
        for (int i = 0; i < NTOPK; ++i) {
            bool active = (!sup[i]) && (s_score[i] > THRESH) && (keep < NMS_KEEP);
            float acc[8] = {0, 0, 0, 0, 0, 0, 0, 0};
            int nm = 0;
            for (int j = 0; j < NTOPK; ++j) {
                if (!sup[j] && s_iou[i * NTOPK + j] > NMS_THRESH) {
                    if (nm < 3)
                        for (int c = 0; c < 8; ++c) acc[c] += s_det[j * 8 + c];
                    nm++;
                }
            }
            int top_n = nm < 3 ? nm : 3;
            float denom = (float)(top_n > 1 ? top_n : 1);
            float vol_i = s_det[i * 8 + 5] * s_det[i * 8 + 6] * s_det[i * 8 + 7];
            bool use_avg = (top_n > 1) && (vol_i > VOL_THRESH);
            float row[8];
            for (int c = 0; c < 8; ++c)
                row[c] = use_avg ? (acc[c] / denom) : s_det[i * 8 + c];
            if (use_avg) { row[0] = 1.0f; row[1] = s_score[i]; }
            if (active) {
                for (int c = 0; c < 8; ++c) s_out[keep * 8 + c] = row[c];
                for (int j = 0; j < NTOPK; ++j)
                    if (!sup[j] && s_iou[i * NTOPK + j] > NMS_THRESH) sup[j] = 1;
                keep++;
            }
        }
    }
    __syncthreads();

    // ---------- Phase 6: write out (B, 60, 8) ----------
    float* ob = out + (size_t)b * NTOPK * 8;
    for (int p = tid; p < NTOPK * 8; p += 256) ob[p] = s_out[p];
}

extern "C" void kernel_launch(void* const* d_in, const int* in_sizes, int n_in,
                              void* d_out, int out_size, void* d_ws, size_t ws_size,
                              hipStream_t stream) {
    const float* cls = (const float*)d_in[0];   // (256,1,24,24,24)
    const float* shp = (const float*)d_in[1];   // (256,3,24,24,24)
    const float* off = (const float*)d_in[2];   // (256,3,24,24,24)
    float* out = (float*)d_out;                 // (256,60,8)
    det_postprocess_kernel<<<dim3(NB), dim3(256), 0, stream>>>(cls, shp, off, out);
}